// IntegratedLaughterModel_6055903887953
// MI455X (gfx1250) — compile-verified
//
#include <hip/hip_runtime.h>
#include <math.h>

#define BB 64
#define SS 2048
#define DD 512
#define HIDD 512
#define NHH 8
#define DHH 64
#define RRR 4

typedef __attribute__((ext_vector_type(2))) float v2f;
typedef __attribute__((ext_vector_type(8))) float v8f;

// ---------------- mask prep: cumsum, split, setup/punch flags ----------------
__global__ void prep_masks(const int* __restrict__ mask, int* __restrict__ flags,
                           float* __restrict__ stats) {
  int b = blockIdx.x;
  if (threadIdx.x != 0) return;
  const int* mrow = mask + (size_t)b * SS;
  int valid = 0;
  for (int s = 0; s < SS; s++) valid += (mrow[s] != 0);
  float splitf = floorf((float)valid * 0.6f);
  int split = (splitf < 1.0f) ? 1 : (int)splitf;
  int punch = valid - split;
  int punch_eff = (punch > 0) ? punch : 1;
  int cum = 0;
  for (int s = 0; s < SS; s++) {
    int m = (mrow[s] != 0);
    cum += m;
    int f = m;
    if (m && cum <= split) f |= 2;
    if (punch > 0) { if (m && cum > split) f |= 4; }
    else           { if (m && cum == valid) f |= 4; }
    flags[(size_t)b * SS + s] = f;
  }
  stats[b * 4 + 0] = (float)valid;
  stats[b * 4 + 1] = (float)split;      // setup count
  stats[b * 4 + 2] = (float)punch_eff;  // punch count (last-token fallback)
}

// ---------------- fold Wk with q_tom: qk[i,h] (padded to 16 heads) ----------------
__global__ void compute_qk(const float* __restrict__ Wk, const float* __restrict__ q_tom,
                           float* __restrict__ qk) {
  int idx = blockIdx.x * blockDim.x + threadIdx.x;
  if (idx >= DD * 16) return;
  int i = idx >> 4, h = idx & 15;
  float acc = 0.f;
  if (h < NHH) {
    for (int d = 0; d < DHH; d++) acc += Wk[(size_t)i * DD + h * DHH + d] * q_tom[h * DHH + d];
  }
  qk[i * 16 + h] = acc;
}

// ---------------- WMMA pass 1: scores = (x @ qk)/8, masked ----------------
__global__ void __launch_bounds__(128) scores_wmma(const float* __restrict__ x,
                                                   const float* __restrict__ qk,
                                                   const int* __restrict__ mask,
                                                   float* __restrict__ attn) {
  __shared__ float qks[DD * 16];
  for (int i = threadIdx.x; i < DD * 16; i += 128) qks[i] = qk[i];
  __syncthreads();
  int wave = threadIdx.x >> 5;
  int lane = threadIdx.x & 31;
  int tile = blockIdx.x * 4 + wave;
  int row0 = tile * 16;                 // 16 tokens per wave; SS%16==0 so no b crossing
  int b = row0 / SS, s0 = row0 % SS;
  int r = lane & 15;
  int koff = (lane >> 4) << 1;          // lanes 16-31 hold K+2
  const float* xrow = x + (size_t)(row0 + r) * DD;
  v8f c = {};
  for (int k0 = 0; k0 < DD; k0 += 4) {
    v2f a = *(const v2f*)(xrow + k0 + koff);                    // A: 16x4 tile of x
    v2f bb;
    bb.x = qks[(k0 + koff) * 16 + r];                           // B: 4x16 tile of qk
    bb.y = qks[(k0 + koff + 1) * 16 + r];
    c = __builtin_amdgcn_wmma_f32_16x16x4_f32(false, a, false, bb, (short)0, c, false, false);
  }
  int head = r;                          // C: N = head, M = token-in-tile
  if (head < NHH) {
    int soff = s0 + ((lane >> 4) << 3);
    const int* mrow = mask + (size_t)b * SS + soff;
    float* orow = attn + ((size_t)(b * NHH + head)) * SS + soff;
    const float scale = 0.125f;          // 1/sqrt(DH)
#pragma unroll
    for (int j = 0; j < 8; j++) {
      float sc = c[j] * scale;
      orow[j] = (mrow[j] != 0) ? sc : -1.0e9f;
    }
  }
}

// ---------------- per (b,h) softmax over S ----------------
__global__ void softmax_rows(float* __restrict__ attn) {
  __shared__ float red[256];
  int row = blockIdx.x;
  float* p = attn + (size_t)row * SS;
  int t = threadIdx.x;
  float mx = -INFINITY;
  for (int s = t; s < SS; s += 256) mx = fmaxf(mx, p[s]);
  red[t] = mx; __syncthreads();
  for (int o = 128; o > 0; o >>= 1) { if (t < o) red[t] = fmaxf(red[t], red[t + o]); __syncthreads(); }
  mx = red[0]; __syncthreads();
  float sum = 0.f;
  for (int s = t; s < SS; s += 256) sum += expf(p[s] - mx);
  red[t] = sum; __syncthreads();
  for (int o = 128; o > 0; o >>= 1) { if (t < o) red[t] += red[t + o]; __syncthreads(); }
  float inv = 1.0f / red[0];
  for (int s = t; s < SS; s += 256) p[s] = expf(p[s] - mx) * inv;
}

// ---------------- masked means: mean_tok, setup_mean|punch_mean into cc ----------------
__global__ void masked_sums(const float* __restrict__ x, const int* __restrict__ flags,
                            const float* __restrict__ stats, float* __restrict__ mean_tok,
                            float* __restrict__ cc) {
  __shared__ int fl[SS];
  int b = blockIdx.x;
  int d = blockIdx.y * 256 + threadIdx.x;
  for (int s = threadIdx.x; s < SS; s += 256) fl[s] = flags[(size_t)b * SS + s];
  __syncthreads();
  float a0 = 0.f, a1 = 0.f, a2 = 0.f;
  const float* xb = x + (size_t)b * SS * DD + d;
  for (int s = 0; s < SS; s++) {
    int f = fl[s];
    if (f) {
      float v = xb[(size_t)s * DD];
      a0 += v;
      if (f & 2) a1 += v;
      if (f & 4) a2 += v;
    }
  }
  mean_tok[(size_t)b * DD + d] = a0 / stats[b * 4 + 0];
  cc[(size_t)b * 2 * DD + d]      = a1 / stats[b * 4 + 1];
  cc[(size_t)b * 2 * DD + DD + d] = a2 / stats[b * 4 + 2];
}

// ---------------- WMMA pass 2: xa[b,h,:] = attn[b,h,:] @ x[b,:,:] ----------------
__global__ void __launch_bounds__(128) xa_wmma(const float* __restrict__ x,
                                               const float* __restrict__ attn,
                                               float* __restrict__ xa) {
  int b = blockIdx.x;
  int wave = threadIdx.x >> 5;
  int lane = threadIdx.x & 31;
  int n0 = (blockIdx.y * 4 + wave) * 16;
  int r = lane & 15;
  int koff = (lane >> 4) << 1;
  const float* arow = (r < NHH) ? (attn + ((size_t)(b * NHH + r)) * SS) : nullptr;
  const float* xb = x + (size_t)b * SS * DD;
  v8f c = {};
  for (int k0 = 0; k0 < SS; k0 += 4) {
    v2f a;
    if (r < NHH) a = *(const v2f*)(arow + k0 + koff);   // A: 16x4 tile of attn (rows 8..15 zero)
    else { a.x = 0.f; a.y = 0.f; }
    v2f bb;                                             // B: 4x16 tile of x
    bb.x = xb[(size_t)(k0 + koff) * DD + n0 + r];
    bb.y = xb[(size_t)(k0 + koff + 1) * DD + n0 + r];
    c = __builtin_amdgcn_wmma_f32_16x16x4_f32(false, a, false, bb, (short)0, c, false, false);
  }
  if (lane < 16) {                                      // lanes 0-15 hold M=0..7 (valid heads)
#pragma unroll
    for (int j = 0; j < 8; j++)
      xa[((size_t)(b * NHH + j)) * DD + n0 + lane] = c[j];
  }
}

// ---------------- fused[b, h*64+dh] = xa[b,h,:] . Wv[:, h*64+dh] ----------------
__global__ void fused_v(const float* __restrict__ xa, const float* __restrict__ Wv,
                        float* __restrict__ fused_flat) {
  int bh = blockIdx.x;
  int b = bh / NHH, h = bh % NHH;
  int dh = threadIdx.x;
  const float* row = xa + (size_t)bh * DD;
  const float* wcol = Wv + h * DHH + dh;
  float acc = 0.f;
  for (int i = 0; i < DD; i++) acc += row[i] * wcol[(size_t)i * DD];
  fused_flat[(size_t)b * DD + h * DHH + dh] = acc;
}

// ---------------- small dense layer: out = act(in @ W + bias) ----------------
__global__ void linear_act(const float* __restrict__ in, int in_stride,
                           const float* __restrict__ W, const float* __restrict__ bias,
                           float* __restrict__ out, int out_stride, int K, int N, int act) {
  int b = blockIdx.x;
  int j = blockIdx.y * blockDim.x + threadIdx.x;
  if (j >= N) return;
  const float* row = in + (size_t)b * in_stride;
  float acc = bias ? bias[j] : 0.f;
  for (int i = 0; i < K; i++) acc += row[i] * W[(size_t)i * N + j];
  if (act) acc = fmaxf(acc, 0.f);
  out[(size_t)b * out_stride + j] = acc;
}

// ---------------- scalar head: out = [sigmoid](in . w + bias) ----------------
__global__ void dot_head(const float* __restrict__ in, int in_stride, int K,
                         const float* __restrict__ w, const float* __restrict__ bias,
                         float* __restrict__ out, int out_stride, int out_off, int mode) {
  __shared__ float red[128];
  int b = blockIdx.x, t = threadIdx.x;
  const float* row = in + (size_t)b * in_stride;
  float acc = 0.f;
  for (int i = t; i < K; i += 128) acc += row[i] * w[i];
  red[t] = acc; __syncthreads();
  for (int o = 64; o > 0; o >>= 1) { if (t < o) red[t] += red[t + o]; __syncthreads(); }
  if (t == 0) {
    float r = red[0] + bias[0];
    if (mode == 1) r = 1.0f / (1.0f + expf(-r));
    out[(size_t)b * out_stride + out_off] = r;
  }
}

// ---------------- mix streams with M=I+UV, normalize, pool, build sev_in ----------------
__global__ void mix_kernel(const float* __restrict__ fm, const float* __restrict__ ctx,
                           const float* __restrict__ cc, const float* __restrict__ U,
                           const float* __restrict__ V, const float* __restrict__ scores3,
                           float* __restrict__ sev_in) {
  __shared__ float red[3][256];
  int b = blockIdx.x, t = threadIdx.x;
  float M3[3][3];
#pragma unroll
  for (int i = 0; i < 3; i++)
#pragma unroll
    for (int j = 0; j < 3; j++) {
      float a = (i == j) ? 1.f : 0.f;
#pragma unroll
      for (int r = 0; r < RRR; r++) a += U[i * RRR + r] * V[r * 3 + j];
      M3[i][j] = a;
    }
  float mx[2][3];
  float sq0 = 0.f, sq1 = 0.f, sq2 = 0.f;
#pragma unroll
  for (int half = 0; half < 2; half++) {
    int d = t + half * 256;
    float s0 = fm[(size_t)b * DD + d];
    float s1 = ctx[(size_t)b * DD + d];
    float s2 = 0.5f * (cc[(size_t)b * 2 * DD + d] + cc[(size_t)b * 2 * DD + DD + d]);
#pragma unroll
    for (int i = 0; i < 3; i++) {
      float v = M3[i][0] * s0 + M3[i][1] * s1 + M3[i][2] * s2;
      mx[half][i] = v;
      if (i == 0) sq0 += v * v; else if (i == 1) sq1 += v * v; else sq2 += v * v;
    }
  }
  red[0][t] = sq0; red[1][t] = sq1; red[2][t] = sq2;
  __syncthreads();
  for (int o = 128; o > 0; o >>= 1) {
    if (t < o) { red[0][t] += red[0][t + o]; red[1][t] += red[1][t + o]; red[2][t] += red[2][t + o]; }
    __syncthreads();
  }
  float n0 = sqrtf(red[0][0]) + 1e-6f;
  float n1 = sqrtf(red[1][0]) + 1e-6f;
  float n2 = sqrtf(red[2][0]) + 1e-6f;
#pragma unroll
  for (int half = 0; half < 2; half++) {
    int d = t + half * 256;
    float pooled = (mx[half][0] / n0 + mx[half][1] / n1 + mx[half][2] / n2) * (1.0f / 3.0f);
    sev_in[(size_t)b * 515 + d] = pooled;
  }
  if (t < 3) sev_in[(size_t)b * 515 + 512 + t] = scores3[b * 3 + t];
}

// ---------------- final combine ----------------
__global__ void final_k(const float* __restrict__ fl, const float* __restrict__ sl,
                        const float* __restrict__ scores3, float* __restrict__ out) {
  int b = threadIdx.x;
  if (b >= BB) return;
  float p = (scores3[b * 3] + scores3[b * 3 + 1] + scores3[b * 3 + 2]) * (1.0f / 3.0f);
  p = fminf(fmaxf(p, 1e-4f), 1.0f - 1e-4f);
  float lg = logf(p / (1.0f - p));
  out[b] = fl[b] + 0.5f * sl[b] + 0.1f * lg;
}

extern "C" void kernel_launch(void* const* d_in, const int* in_sizes, int n_in,
                              void* d_out, int out_size, void* d_ws, size_t ws_size,
                              hipStream_t stream) {
  const float* x     = (const float*)d_in[0];
  const int*   mask  = (const int*)d_in[1];
  const float* Wk    = (const float*)d_in[2];
  const float* Wv    = (const float*)d_in[3];
  const float* q_tom = (const float*)d_in[4];
  const float* Wtf   = (const float*)d_in[5];
  const float* w_hp  = (const float*)d_in[6];
  const float* b_hp  = (const float*)d_in[7];
  const float* Wg1   = (const float*)d_in[8];
  const float* bg1   = (const float*)d_in[9];
  const float* Wg2   = (const float*)d_in[10];
  const float* bg2   = (const float*)d_in[11];
  const float* w_inc = (const float*)d_in[12];
  const float* b_inc = (const float*)d_in[13];
  const float* Wc1   = (const float*)d_in[14];
  const float* bc1   = (const float*)d_in[15];
  const float* wc2   = (const float*)d_in[16];
  const float* bc2   = (const float*)d_in[17];
  const float* U     = (const float*)d_in[18];
  const float* V     = (const float*)d_in[19];
  const float* Ws1   = (const float*)d_in[20];
  const float* bs1   = (const float*)d_in[21];
  const float* ws2   = (const float*)d_in[22];
  const float* bs2   = (const float*)d_in[23];
  const float* Wf1   = (const float*)d_in[24];
  const float* bf1   = (const float*)d_in[25];
  const float* wf2   = (const float*)d_in[26];
  const float* bf2   = (const float*)d_in[27];
  float* out = (float*)d_out;

  char* p = (char*)d_ws;
  auto alloc = [&](size_t nfloats) {
    float* r = (float*)p;
    p += ((nfloats * 4 + 255) / 256) * 256;
    return r;
  };
  float* attn     = alloc((size_t)BB * NHH * SS);   // scores -> attn in place
  int*   flags    = (int*)alloc((size_t)BB * SS);
  float* stats    = alloc(BB * 4);
  float* qk       = alloc(DD * 16);
  float* mean_tok = alloc((size_t)BB * DD);
  float* cc       = alloc((size_t)BB * 2 * DD);     // [setup_mean | punch_mean]
  float* xa       = alloc((size_t)BB * NHH * DD);
  float* ff       = alloc((size_t)BB * DD);         // fused (B,D)
  float* fm       = alloc((size_t)BB * DD);         // fused_mental
  float* gh       = alloc((size_t)BB * HIDD);
  float* ctx      = alloc((size_t)BB * DD);
  float* ch       = alloc((size_t)BB * HIDD);
  float* scores3  = alloc(BB * 3);
  float* sev_in   = alloc((size_t)BB * 515);
  float* hs       = alloc((size_t)BB * HIDD);
  float* hf       = alloc((size_t)BB * HIDD);
  float* sl       = alloc(BB);
  float* fl       = alloc(BB);

  prep_masks<<<BB, 32, 0, stream>>>(mask, flags, stats);
  compute_qk<<<(DD * 16 + 255) / 256, 256, 0, stream>>>(Wk, q_tom, qk);
  scores_wmma<<<(BB * SS) / 64, 128, 0, stream>>>(x, qk, mask, attn);
  softmax_rows<<<BB * NHH, 256, 0, stream>>>(attn);
  masked_sums<<<dim3(BB, DD / 256), 256, 0, stream>>>(x, flags, stats, mean_tok, cc);
  xa_wmma<<<dim3(BB, (DD / 16) / 4), 128, 0, stream>>>(x, attn, xa);
  fused_v<<<BB * NHH, DHH, 0, stream>>>(xa, Wv, ff);
  linear_act<<<dim3(BB, DD / 256), 256, 0, stream>>>(ff, DD, Wtf, nullptr, fm, DD, DD, DD, 0);
  dot_head<<<BB, 128, 0, stream>>>(fm, DD, DD, w_hp, b_hp, scores3, 3, 0, 1);
  linear_act<<<dim3(BB, HIDD / 256), 256, 0, stream>>>(mean_tok, DD, Wg1, bg1, gh, HIDD, DD, HIDD, 1);
  linear_act<<<dim3(BB, DD / 256), 256, 0, stream>>>(gh, HIDD, Wg2, bg2, ctx, DD, HIDD, DD, 0);
  dot_head<<<BB, 128, 0, stream>>>(ctx, DD, DD, w_inc, b_inc, scores3, 3, 1, 1);
  linear_act<<<dim3(BB, HIDD / 256), 256, 0, stream>>>(cc, 2 * DD, Wc1, bc1, ch, HIDD, 2 * DD, HIDD, 1);
  dot_head<<<BB, 128, 0, stream>>>(ch, HIDD, HIDD, wc2, bc2, scores3, 3, 2, 1);
  mix_kernel<<<BB, 256, 0, stream>>>(fm, ctx, cc, U, V, scores3, sev_in);
  linear_act<<<dim3(BB, HIDD / 256), 256, 0, stream>>>(sev_in, 515, Ws1, bs1, hs, HIDD, 515, HIDD, 1);
  dot_head<<<BB, 128, 0, stream>>>(hs, HIDD, HIDD, ws2, bs2, sl, 1, 0, 0);
  linear_act<<<dim3(BB, HIDD / 256), 256, 0, stream>>>(sev_in, 515, Wf1, bf1, hf, HIDD, 515, HIDD, 1);
  dot_head<<<BB, 128, 0, stream>>>(hf, HIDD, HIDD, wf2, bf2, fl, 1, 0, 0);
  final_k<<<1, 64, 0, stream>>>(fl, sl, scores3, out);
}